// RNN_74096775790863
// MI455X (gfx1250) — compile-verified
//
#include <hip/hip_runtime.h>

// ---------------------------------------------------------------------------
// 2-layer LSTM scan (T=512) fused with final projection, for MI455X (gfx1250).
// B=1024 split across 64 workgroups of 16 batch rows (WMMA M-tile = 16).
// Per step:  G1 = [e|h1] @ W1   (K=85 -> 22 k-tiles of V_WMMA_F32_16X16X4_F32)
//            gates -> h1,c1 ; a = relu(h1)
//            G2 = [a|h2] @ W2   (K=105 -> 27 k-tiles)
//            gates -> h2,c2 ; logits += h2 @ Wout[t]
// Weights pre-swizzled into LDS ([ktile][n][k%4]) so A and B WMMA fragments
// are each one ds_load_b64 per lane. fp32 WMMA keeps reference precision.
// Two independent K-accumulation chains per N-tile halve the WMMA RAW chain.
// ---------------------------------------------------------------------------

typedef float v2f __attribute__((ext_vector_type(2)));
typedef float v8f __attribute__((ext_vector_type(8)));

#define EMBD    10
#define H1D     75
#define H2D     30
#define TSTEPS  512

#define K1      85            // EMB + H1
#define K2      105           // H1 + H2
#define KT1     22            // ceil(K1/4)
#define KT2     27            // ceil(K2/4)
#define N1      300           // 4*H1
#define N2      120           // 4*H2
#define N1P     304           // 19 N-tiles
#define N2P     128           // 8 N-tiles
#define NT1     19
#define NT2     8
#define AP1     90            // A1 LDS pitch (>= KT1*4, even)
#define AP2     110           // A2 LDS pitch (>= KT2*4, even)
#define C1P     76
#define C2P     32

#define W1S_F   (KT1 * N1P * 4)   // 26752 floats
#define W2S_F   (KT2 * N2P * 4)   // 13824
#define A1S_F   (16 * AP1)        // 1440
#define A2S_F   (16 * AP2)        // 1760
#define G1S_F   (16 * N1P)        // 4864
#define G2S_F   (16 * N2P)        // 2048
#define C1S_F   (16 * C1P)        // 1216
#define C2S_F   (16 * C2P)        // 512
#define LDS_F   (W1S_F + W2S_F + A1S_F + A2S_F + G1S_F + G2S_F + C1S_F + C2S_F)

__device__ __forceinline__ float fast_sigmoid(float x) {
    return 1.0f / (1.0f + __expf(-x));
}
__device__ __forceinline__ float fast_tanh(float x) {
    float xc = fminf(fmaxf(x, -15.0f), 15.0f);
    float e  = __expf(2.0f * xc);
    return (e - 1.0f) / (e + 1.0f);
}

__device__ __forceinline__ v8f wmma_f32(v2f a, v2f b, v8f c) {
    // V_WMMA_F32_16X16X4_F32: D = A(16x4) * B(4x16) + C(16x16)
    return __builtin_amdgcn_wmma_f32_16x16x4_f32(
        /*neg_a=*/false, a, /*neg_b=*/false, b,
        /*c_mod=*/(short)0, c, /*reuse_a=*/false, /*reuse_b=*/false);
}

__global__ void __launch_bounds__(256, 1)
lstm_scan_kernel(const int*   __restrict__ xidx,   // [B, T]
                 const float* __restrict__ embed,  // [VOCAB, EMB]
                 const float* __restrict__ W1,     // [85, 300]
                 const float* __restrict__ b1,     // [300]
                 const float* __restrict__ W2,     // [105, 120]
                 const float* __restrict__ b2,     // [120]
                 const float* __restrict__ Wout,   // [T*H2, 2]
                 const float* __restrict__ bout,   // [2]
                 float*       __restrict__ out)    // [B, 2]
{
    extern __shared__ float smem[];
    float* W1s = smem;                 // packed [kt][n][kr]
    float* W2s = W1s + W1S_F;
    float* A1  = W2s + W2S_F;          // [16][AP1]: cols 0..9 = e, 10..84 = h1
    float* A2  = A1 + A1S_F;           // [16][AP2]: cols 0..74 = relu(h1), 75..104 = h2
    float* G1  = A2 + A2S_F;           // [16][N1P]
    float* G2  = G1 + G1S_F;           // [16][N2P]
    float* C1s = G2 + G2S_F;           // [16][C1P]
    float* C2s = C1s + C1S_F;          // [16][C2P]

    const int tid  = threadIdx.x;
    const int wave = tid >> 5;
    const int lane = tid & 31;
    const int ln   = lane & 15;               // N-lane within tile
    const int kk   = (lane < 16) ? 0 : 2;     // K sub-offset for A/B fragments
    const int bm0  = blockIdx.x * 16;         // batch-row base for this block

    // ---- stage W1 into LDS, swizzled to [kt][n][kr], zero-padded ----
    for (int idx = tid; idx < W1S_F; idx += 256) {
        int kt = idx / (N1P * 4);
        int rem = idx - kt * (N1P * 4);
        int n  = rem >> 2;
        int kr = rem & 3;
        int k  = kt * 4 + kr;
        W1s[idx] = (k < K1 && n < N1) ? W1[k * N1 + n] : 0.0f;
    }
    // ---- stage W2 ----
    for (int idx = tid; idx < W2S_F; idx += 256) {
        int kt = idx / (N2P * 4);
        int rem = idx - kt * (N2P * 4);
        int n  = rem >> 2;
        int kr = rem & 3;
        int k  = kt * 4 + kr;
        W2s[idx] = (k < K2 && n < N2) ? W2[k * N2 + n] : 0.0f;
    }
    // ---- zero activation panels A1, A2 (h starts at 0; pads stay 0) and head of G1; C1s/C2s not zeroed ----
    for (int idx = tid; idx < A1S_F + A2S_F + C1S_F + C2S_F; idx += 256)
        A1[idx] = 0.0f;
    __syncthreads();

    float acc = 0.0f;   // logits accumulator (valid for tid < 32)

    for (int t = 0; t < TSTEPS; ++t) {
        // ---- embedding gather: A1[m][0..9] = embed[x[bm0+m, t]] ----
        if (tid < 16 * EMBD) {
            int m = tid / EMBD, e = tid - m * EMBD;
            int tok = xidx[(bm0 + m) * TSTEPS + t];
            A1[m * AP1 + e] = embed[tok * EMBD + e];
        }
        __syncthreads();

        // ---- GEMM1: [16 x 85] @ [85 x 300] -> G1, 19 N-tiles over 8 waves ----
        for (int nt = wave; nt < NT1; nt += 8) {
            v8f ca = {}; v8f cb = {};
            const float* ap = A1 + ln * AP1 + kk;
            const float* bp = W1s + (nt * 16 + ln) * 4 + kk;
            #pragma unroll
            for (int kt = 0; kt < KT1; kt += 2) {        // 2 chains of 11
                v2f a0 = *(const v2f*)(ap);
                v2f b0 = *(const v2f*)(bp);
                v2f a1 = *(const v2f*)(ap + 4);
                v2f b1v = *(const v2f*)(bp + N1P * 4);
                ca = wmma_f32(a0, b0, ca);
                cb = wmma_f32(a1, b1v, cb);
                ap += 8;
                bp += N1P * 8;
            }
            v8f c = ca + cb;
            int mbase = (lane >> 4) * 8;
            #pragma unroll
            for (int r = 0; r < 8; ++r)
                G1[(mbase + r) * N1P + nt * 16 + ln] = c[r];
        }
        __syncthreads();

        // ---- LSTM1 gates -> h1, c1; A2[:,0:75] = relu(h1) ----
        for (int idx = tid; idx < 16 * H1D; idx += 256) {
            int m = idx / H1D, j = idx - m * H1D;
            const float* g = G1 + m * N1P;
            float iv = g[j]            + b1[j];
            float gv = g[H1D + j]      + b1[H1D + j];
            float fv = g[2 * H1D + j]  + b1[2 * H1D + j];
            float ov = g[3 * H1D + j]  + b1[3 * H1D + j];
            float f  = fast_sigmoid(fv + 1.0f);
            float c  = f * C1s[m * C1P + j] + fast_sigmoid(iv) * fast_tanh(gv);
            float h  = fast_sigmoid(ov) * fast_tanh(c);
            C1s[m * C1P + j]   = c;
            A1[m * AP1 + EMBD + j] = h;
            A2[m * AP2 + j]        = fmaxf(h, 0.0f);
        }
        __syncthreads();

        // ---- GEMM2: [16 x 105] @ [105 x 120] -> G2, 8 N-tiles / 8 waves ----
        {
            int nt = wave;
            v8f ca = {}; v8f cb = {};
            const float* ap = A2 + ln * AP2 + kk;
            const float* bp = W2s + (nt * 16 + ln) * 4 + kk;
            #pragma unroll
            for (int kt = 0; kt + 1 < KT2; kt += 2) {    // 26 tiles paired
                v2f a0 = *(const v2f*)(ap);
                v2f b0 = *(const v2f*)(bp);
                v2f a1 = *(const v2f*)(ap + 4);
                v2f b1v = *(const v2f*)(bp + N2P * 4);
                ca = wmma_f32(a0, b0, ca);
                cb = wmma_f32(a1, b1v, cb);
                ap += 8;
                bp += N2P * 8;
            }
            {   // final odd k-tile (kt = 26)
                v2f a0 = *(const v2f*)(ap);
                v2f b0 = *(const v2f*)(bp);
                ca = wmma_f32(a0, b0, ca);
            }
            v8f c = ca + cb;
            int mbase = (lane >> 4) * 8;
            #pragma unroll
            for (int r = 0; r < 8; ++r)
                G2[(mbase + r) * N2P + nt * 16 + ln] = c[r];
        }
        __syncthreads();

        // ---- LSTM2 gates -> h2, c2; A2[:,75:105] = h2 ----
        for (int idx = tid; idx < 16 * H2D; idx += 256) {
            int m = idx / H2D, j = idx - m * H2D;
            const float* g = G2 + m * N2P;
            float iv = g[j]            + b2[j];
            float gv = g[H2D + j]      + b2[H2D + j];
            float fv = g[2 * H2D + j]  + b2[2 * H2D + j];
            float ov = g[3 * H2D + j]  + b2[3 * H2D + j];
            float f  = fast_sigmoid(fv + 1.0f);
            float c  = f * C2s[m * C2P + j] + fast_sigmoid(iv) * fast_tanh(gv);
            float h  = fast_sigmoid(ov) * fast_tanh(c);
            C2s[m * C2P + j]       = c;
            A2[m * AP2 + H1D + j]  = h;
        }
        __syncthreads();

        // ---- fused output projection: logits[m][col] += h2[m,:] @ Wout[t] ----
        if (tid < 32) {
            int m = tid >> 1, col = tid & 1;
            const float* wo = Wout + (t * H2D) * 2 + col;
            float s = acc;
            #pragma unroll
            for (int j = 0; j < H2D; ++j)
                s += A2[m * AP2 + H1D + j] * wo[j * 2];
            acc = s;
        }
        // no barrier needed here: next writers of A1[0:10]/A2[75:105] are
        // separated from these readers by >=1 barrier in the next iteration
    }

    if (tid < 32) {
        int m = tid >> 1, col = tid & 1;
        out[(bm0 + m) * 2 + col] = acc + bout[col];
    }
}

extern "C" void kernel_launch(void* const* d_in, const int* in_sizes, int n_in,
                              void* d_out, int out_size, void* d_ws, size_t ws_size,
                              hipStream_t stream) {
    const int*   x     = (const int*)  d_in[0];
    const float* embed = (const float*)d_in[1];
    const float* W1    = (const float*)d_in[2];
    const float* b1    = (const float*)d_in[3];
    const float* W2    = (const float*)d_in[4];
    const float* b2    = (const float*)d_in[5];
    const float* Wout  = (const float*)d_in[6];
    const float* bout  = (const float*)d_in[7];
    float* out = (float*)d_out;

    // allow >64KB dynamic LDS (gfx1250 WGP has 320KB). Host-side attribute,
    // idempotent and deterministic (same call every invocation), not a stream
    // op, so graph-capture safe. No static guards per harness contract.
    hipFuncSetAttribute((const void*)lstm_scan_kernel,
                        hipFuncAttributeMaxDynamicSharedMemorySize,
                        LDS_F * sizeof(float));

    hipLaunchKernelGGL(lstm_scan_kernel, dim3(1024 / 16), dim3(256),
                       LDS_F * sizeof(float), stream,
                       x, embed, W1, b1, W2, b2, Wout, bout, out);
}